// PhysicsInformedNN_28501402976478
// MI455X (gfx1250) — compile-verified
//
#include <hip/hip_runtime.h>
#include <hip/hip_bf16.h>

typedef float v2f __attribute__((ext_vector_type(2)));
typedef float v8f __attribute__((ext_vector_type(8)));

#define NPTW 16            // points per wave
#define NJET 13            // jet components per scalar
#define ROWS (NPTW * NJET) // 208 rows per wave-block
#define HID 20
#define NHL 7

// Jet index map: 0=val 1=dx 2=dy 3=dt 4=dxx 5=dxy 6=dyy 7=dxt 8=dyt
//                9=dxxx 10=dxxy 11=dxyy 12=dyyy

__device__ __forceinline__ float fast_tanh(float v) {
#if __has_builtin(__builtin_amdgcn_tanhf)
    return __builtin_amdgcn_tanhf(v);   // CDNA5 V_TANH_F32 (TRANS pipe)
#else
    return tanhf(v);
#endif
}

__device__ __forceinline__ void tanh_jet(const float* u, float* o) {
    float f0 = fast_tanh(u[0]);
    float f1 = 1.0f - f0 * f0;           // f'
    float f2 = -2.0f * f0 * f1;          // f''
    float f3 = 4.0f * f0 * f0 * f1 - 2.0f * f1 * f1; // f'''
    o[0] = f0;
    o[1] = f1 * u[1];
    o[2] = f1 * u[2];
    o[3] = f1 * u[3];
    o[4] = f1 * u[4] + f2 * u[1] * u[1];
    o[5] = f1 * u[5] + f2 * u[1] * u[2];
    o[6] = f1 * u[6] + f2 * u[2] * u[2];
    o[7] = f1 * u[7] + f2 * u[1] * u[3];
    o[8] = f1 * u[8] + f2 * u[2] * u[3];
    o[9]  = f1 * u[9]  + 3.0f * f2 * u[1] * u[4] + f3 * u[1] * u[1] * u[1];
    o[10] = f1 * u[10] + f2 * (u[2] * u[4] + 2.0f * u[1] * u[5]) + f3 * u[1] * u[1] * u[2];
    o[11] = f1 * u[11] + f2 * (u[1] * u[6] + 2.0f * u[2] * u[5]) + f3 * u[1] * u[2] * u[2];
    o[12] = f1 * u[12] + 3.0f * f2 * u[2] * u[6] + f3 * u[2] * u[2] * u[2];
}

__global__ __launch_bounds__(32) void pinn_ns_wmma_kernel(
    const float* __restrict__ x, const float* __restrict__ y,
    const float* __restrict__ t,
    const float* __restrict__ W_in, const float* __restrict__ b_in,
    const float* __restrict__ W_hid, const float* __restrict__ b_hid,
    const float* __restrict__ W_out, const float* __restrict__ b_out,
    const float* __restrict__ lam1, const float* __restrict__ lam2,
    float* __restrict__ out, int Npts)
{
    // Hidden weights pre-swizzled into exact WMMA B-operand lane layout:
    // sWb[l][kt][nt][lane] = { W[kb][n], W[kb+1][n] },
    //   kb = kt*4 + 2*(lane>>4), n = nt*16 + (lane&15)  (zero-padded for n>=20)
    __shared__ float sWb[NHL][5][2][32][2];
    __shared__ float sBh[NHL][HID];
    __shared__ float sWin[3][HID];
    __shared__ float sBin[HID];
    __shared__ float sWout[HID][2];
    __shared__ float sBout[2];
    __shared__ float sAct[2][ROWS][HID];  // ping-pong jet activations
    __shared__ float sx[NPTW], sy[NPTW], st[NPTW];
    __shared__ float sJet[NPTW][2][NJET];

    const int lane = threadIdx.x;
    const int ll = lane & 15;   // low lane id
    const int lh = lane >> 4;   // lane half (0/1)
    const int gbase = blockIdx.x * NPTW;

    // ---- load weights / inputs into LDS ----
    {
        const int n = ll;                    // this lane's B column (per nt offset below)
        #pragma unroll
        for (int l = 0; l < NHL; ++l)
            #pragma unroll
            for (int kt = 0; kt < 5; ++kt) {
                int kb = kt * 4 + 2 * lh;
                #pragma unroll
                for (int nt = 0; nt < 2; ++nt) {
                    int col = nt * 16 + n;
                    float w0 = 0.0f, w1 = 0.0f;
                    if (col < HID) {
                        w0 = W_hid[(l * HID + kb + 0) * HID + col];
                        w1 = W_hid[(l * HID + kb + 1) * HID + col];
                    }
                    sWb[l][kt][nt][lane][0] = w0;
                    sWb[l][kt][nt][lane][1] = w1;
                }
            }
    }
    for (int i = lane; i < NHL * HID; i += 32) sBh[i / HID][i % HID] = b_hid[i];
    for (int i = lane; i < 3 * HID; i += 32) sWin[i / HID][i % HID] = W_in[i];
    for (int i = lane; i < HID; i += 32) sBin[i] = b_in[i];
    for (int i = lane; i < HID * 2; i += 32) sWout[i >> 1][i & 1] = W_out[i];
    if (lane < 2) sBout[lane] = b_out[lane];
    if (lane < NPTW) {
        int gp = gbase + lane;
        bool ok = gp < Npts;
        sx[lane] = ok ? x[gp] : 0.0f;
        sy[lane] = ok ? y[gp] : 0.0f;
        st[lane] = ok ? t[gp] : 0.0f;
    }
    __syncthreads();

    // ---- input layer (jets of inputs are sparse: only value + unit first-order) ----
    for (int pr = lane; pr < NPTW * HID; pr += 32) {
        int pt = pr / HID, un = pr % HID;
        float w0 = sWin[0][un], w1 = sWin[1][un], w2 = sWin[2][un];
        float u0 = sx[pt] * w0 + sy[pt] * w1 + st[pt] * w2 + sBin[un];
        float f0 = fast_tanh(u0);
        float f1 = 1.0f - f0 * f0;
        float f2 = -2.0f * f0 * f1;
        float f3 = 4.0f * f0 * f0 * f1 - 2.0f * f1 * f1;
        float o[NJET];
        o[0] = f0;
        o[1] = f1 * w0; o[2] = f1 * w1; o[3] = f1 * w2;
        o[4] = f2 * w0 * w0; o[5] = f2 * w0 * w1; o[6] = f2 * w1 * w1;
        o[7] = f2 * w0 * w2; o[8] = f2 * w1 * w2;
        o[9]  = f3 * w0 * w0 * w0;
        o[10] = f3 * w0 * w0 * w1;
        o[11] = f3 * w0 * w1 * w1;
        o[12] = f3 * w1 * w1 * w1;
        #pragma unroll
        for (int j = 0; j < NJET; ++j) sAct[0][pt * NJET + j][un] = o[j];
    }
    __syncthreads();

    // ---- 7 hidden layers: jet GEMM (WMMA f32 16x16x4) + tanh jet composition ----
    for (int l = 0; l < NHL; ++l) {
        const float (*in)[HID] = sAct[l & 1];
        float (*ob)[HID] = sAct[(l & 1) ^ 1];

        // B operands: one conflict-free ds_load_b64 per (kt, nt), already in lane layout
        v2f bt[5][2];
        #pragma unroll
        for (int kt = 0; kt < 5; ++kt)
            #pragma unroll
            for (int nt = 0; nt < 2; ++nt)
                bt[kt][nt] = *(const v2f*)&sWb[l][kt][nt][lane][0];

        for (int mt = 0; mt < NJET; ++mt) {   // 13 M-tiles of 16 rows
            v2f at[5];
            #pragma unroll
            for (int kt = 0; kt < 5; ++kt)    // A: lane ll = row M, lh selects K pair
                at[kt] = *(const v2f*)&in[mt * 16 + ll][kt * 4 + 2 * lh];
            #pragma unroll
            for (int nt = 0; nt < 2; ++nt) {
                v8f c = {0.f, 0.f, 0.f, 0.f, 0.f, 0.f, 0.f, 0.f};
                #pragma unroll
                for (int kt = 0; kt < 5; ++kt)
                    c = __builtin_amdgcn_wmma_f32_16x16x4_f32(
                            false, at[kt], false, bt[kt][nt], 0, c, false, false);
                int col = nt * 16 + ll;
                if (col < HID) {
                    #pragma unroll
                    for (int v = 0; v < 8; ++v)    // D: VGPR v -> row v (+8 for high half)
                        ob[mt * 16 + v + 8 * lh][col] = c[v];
                }
            }
        }
        __syncthreads();

        // tanh jet composition, in place on ob (bias applies to value jet only)
        for (int pr = lane; pr < NPTW * HID; pr += 32) {
            int pt = pr / HID, un = pr % HID;
            float u[NJET], o[NJET];
            #pragma unroll
            for (int j = 0; j < NJET; ++j) u[j] = ob[pt * NJET + j][un];
            u[0] += sBh[l][un];
            tanh_jet(u, o);
            #pragma unroll
            for (int j = 0; j < NJET; ++j) ob[pt * NJET + j][un] = o[j];
        }
        __syncthreads();
    }

    // ---- output layer (20 -> 2), one lane per (point, output) ----
    {
        int pt = ll, oidx = lh;
        float jet[NJET];
        #pragma unroll
        for (int j = 0; j < NJET; ++j) jet[j] = 0.0f;
        for (int uu = 0; uu < HID; ++uu) {
            float w = sWout[uu][oidx];
            #pragma unroll
            for (int j = 0; j < NJET; ++j)
                jet[j] += sAct[1][pt * NJET + j][uu] * w;   // final acts live in sAct[1]
        }
        jet[0] += sBout[oidx];
        #pragma unroll
        for (int j = 0; j < NJET; ++j) sJet[pt][oidx][j] = jet[j];
    }
    __syncthreads();

    // ---- residual assembly, one lane per point ----
    if (lane < NPTW) {
        int gp = gbase + lane;
        if (gp < Npts) {
            const float* ps = sJet[lane][0];  // psi jets
            const float* pj = sJet[lane][1];  // p jets
            float uu = ps[2];          // u = psi_y
            float vv = -ps[1];         // v = -psi_x
            float pp = pj[0];
            float u_t = ps[8], u_x = ps[5], u_y = ps[6];
            float u_xx = ps[10], u_yy = ps[12];
            float v_t = -ps[7], v_x = -ps[4], v_y = -ps[5];
            float v_xx = -ps[9], v_yy = -ps[11];
            float p_x = pj[1], p_y = pj[2];
            float l1 = lam1[0], l2 = lam2[0];
            float f_u = u_t + l1 * (uu * u_x + vv * u_y) + p_x - l2 * (u_xx + u_yy);
            float f_v = v_t + l1 * (uu * v_x + vv * v_y) + p_y - l2 * (v_xx + v_yy);
            out[0 * Npts + gp] = uu;
            out[1 * Npts + gp] = vv;
            out[2 * Npts + gp] = pp;
            out[3 * Npts + gp] = f_u;
            out[4 * Npts + gp] = f_v;
        }
    }
}

extern "C" void kernel_launch(void* const* d_in, const int* in_sizes, int n_in,
                              void* d_out, int out_size, void* d_ws, size_t ws_size,
                              hipStream_t stream) {
    (void)n_in; (void)d_ws; (void)ws_size; (void)out_size;
    const float* x     = (const float*)d_in[0];
    const float* y     = (const float*)d_in[1];
    const float* t     = (const float*)d_in[2];
    const float* W_in  = (const float*)d_in[3];
    const float* b_in  = (const float*)d_in[4];
    const float* W_hid = (const float*)d_in[5];
    const float* b_hid = (const float*)d_in[6];
    const float* W_out = (const float*)d_in[7];
    const float* b_out = (const float*)d_in[8];
    const float* lam1  = (const float*)d_in[9];
    const float* lam2  = (const float*)d_in[10];
    float* out = (float*)d_out;
    int Npts = in_sizes[0];
    int nblocks = (Npts + NPTW - 1) / NPTW;
    pinn_ns_wmma_kernel<<<nblocks, 32, 0, stream>>>(
        x, y, t, W_in, b_in, W_hid, b_hid, W_out, b_out, lam1, lam2, out, Npts);
}